// NTXentLoss_20126216749996
// MI455X (gfx1250) — compile-verified
//
#include <hip/hip_runtime.h>

typedef float v2f __attribute__((ext_vector_type(2)));
typedef float v8f __attribute__((ext_vector_type(8)));

#define B_   2048
#define P_   16
#define N_   128
#define D_   512
#define TEMP_INV 10.0f     // 1 / 0.1
#define EPS_ 1e-8f

// One block per anchor b. 8 waves x 32 lanes. Wave w computes dot(neg_row, target)
// for rows 16w..16w+15 via V_WMMA_F32_16X16X4_F32 (B = target chunk broadcast to
// all 16 columns -> every column of D holds the 16 dots). Row norms accumulate in
// VALU alongside. Then masked LSE over [pos_sim, neg_sims].
__global__ __launch_bounds__(256) void ntxent_main_kernel(
    const float* __restrict__ target,
    const float* __restrict__ positives,
    const float* __restrict__ negatives,
    const int* __restrict__ pos_idx,
    const unsigned char* __restrict__ neg_mask,
    float* __restrict__ block_loss)
{
    __shared__ float t_lds[D_];
    __shared__ float red[3][8];
    __shared__ float dot_s[N_];
    __shared__ float sims[N_];

    const int b    = blockIdx.x;
    const int tid  = threadIdx.x;
    const int lane = tid & 31;
    const int wave = tid >> 5;

    // ---- Preamble: stage target to LDS; block-reduce ||t||^2, pos.t, ||pos||^2
    const float* tb = target + (size_t)b * D_;
    const float* pb = positives + ((size_t)b * P_ + (size_t)pos_idx[b]) * D_;

    float2 tv = *(const float2*)(tb + tid * 2);
    float2 pv = *(const float2*)(pb + tid * 2);
    *(float2*)(&t_lds[tid * 2]) = tv;

    float tsq  = tv.x * tv.x + tv.y * tv.y;
    float pdot = pv.x * tv.x + pv.y * tv.y;
    float psq  = pv.x * pv.x + pv.y * pv.y;
    #pragma unroll
    for (int off = 16; off; off >>= 1) {
        tsq  += __shfl_xor(tsq,  off, 32);
        pdot += __shfl_xor(pdot, off, 32);
        psq  += __shfl_xor(psq,  off, 32);
    }
    if (lane == 0) { red[0][wave] = tsq; red[1][wave] = pdot; red[2][wave] = psq; }
    __syncthreads();

    float tsq_t = 0.f, pdot_t = 0.f, psq_t = 0.f;
    #pragma unroll
    for (int i = 0; i < 8; ++i) {
        tsq_t += red[0][i]; pdot_t += red[1][i]; psq_t += red[2][i];
    }

    // ---- Main WMMA loop: 16 negatives per wave, K marches over D in steps of 4.
    // A frag (ISA 16x4 f32 layout): lane l<16 -> row l, k={0,1}; lane l+16 -> row l-16, k={2,3}
    const int row = lane & 15;
    const int h   = lane >> 4;               // 0 for k={0,1}, 1 for k={2,3}
    const float* aptr = negatives + ((size_t)b * N_ + (size_t)(wave * 16 + row)) * D_ + 2 * h;
    const float* bptr = t_lds + 2 * h;       // all 16 B-columns identical (broadcast)

    v8f   c   = {0.f, 0.f, 0.f, 0.f, 0.f, 0.f, 0.f, 0.f};
    float nsq = 0.f;

    #pragma unroll 4
    for (int d = 0; d < D_; d += 4) {
        v2f a  = *(const v2f*)(aptr + d);    // global b64, rows 2KB apart, line-reuse across iters
        v2f bf = *(const v2f*)(bptr + d);    // LDS b64 broadcast
        c = __builtin_amdgcn_wmma_f32_16x16x4_f32(
                /*neg_a=*/false, a, /*neg_b=*/false, bf,
                /*c_mod=*/(short)0, c, /*reuse_a=*/false, /*reuse_b=*/false);
        nsq = fmaf(a.x, a.x, fmaf(a.y, a.y, nsq));
    }
    // combine the two k-half partial norms: lanes l and l+16 hold halves of row l
    nsq += __shfl_xor(nsq, 16, 32);

    // ---- Extract dots: C/D layout: VGPR r, lanes 0-15 -> M=r, lanes 16-31 -> M=r+8;
    // all N columns identical, so lane 0 / lane 16 suffice. Same-wave DS ops are in-order.
    if (lane == 0) {
        #pragma unroll
        for (int r = 0; r < 8; ++r) dot_s[wave * 16 + r] = c[r];
    } else if (lane == 16) {
        #pragma unroll
        for (int r = 0; r < 8; ++r) dot_s[wave * 16 + 8 + r] = c[r];
    }

    if (lane < 16) {
        int rg = wave * 16 + lane;
        float dot   = dot_s[rg];
        float denom = fmaxf(sqrtf(nsq * tsq_t), EPS_);   // ||neg||*||t||
        float sim   = dot / denom * TEMP_INV;
        bool  m     = neg_mask[(size_t)b * N_ + rg] != 0;
        sims[rg]    = m ? sim : -__builtin_inff();
    }
    __syncthreads();

    // ---- Masked logsumexp over [pos_sim, sims[0..127]] by wave 0
    if (wave == 0) {
        float pos_sim = pdot_t / fmaxf(sqrtf(psq_t * tsq_t), EPS_) * TEMP_INV;
        float x0 = sims[lane], x1 = sims[lane + 32], x2 = sims[lane + 64], x3 = sims[lane + 96];
        float mx = fmaxf(fmaxf(x0, x1), fmaxf(x2, x3));
        mx = fmaxf(mx, pos_sim);
        #pragma unroll
        for (int off = 16; off; off >>= 1) mx = fmaxf(mx, __shfl_xor(mx, off, 32));
        float s = expf(x0 - mx) + expf(x1 - mx) + expf(x2 - mx) + expf(x3 - mx);
        if (lane == 0) s += expf(pos_sim - mx);
        #pragma unroll
        for (int off = 16; off; off >>= 1) s += __shfl_xor(s, off, 32);
        if (lane == 0) block_loss[b] = (logf(s) + mx) - pos_sim;
    }
}

// Deterministic fixed-order mean over the 2048 per-block losses.
__global__ __launch_bounds__(256) void ntxent_reduce_kernel(
    const float* __restrict__ block_loss, float* __restrict__ out)
{
    __shared__ float r[256];
    int tid = threadIdx.x;
    float s = 0.f;
    #pragma unroll
    for (int i = tid; i < B_; i += 256) s += block_loss[i];
    r[tid] = s;
    __syncthreads();
    #pragma unroll
    for (int off = 128; off; off >>= 1) {
        if (tid < off) r[tid] += r[tid + off];
        __syncthreads();
    }
    if (tid == 0) out[0] = r[0] * (1.0f / (float)B_);
}

extern "C" void kernel_launch(void* const* d_in, const int* in_sizes, int n_in,
                              void* d_out, int out_size, void* d_ws, size_t ws_size,
                              hipStream_t stream) {
    (void)in_sizes; (void)n_in; (void)out_size; (void)ws_size;
    const float*         target    = (const float*)d_in[0];
    const float*         positives = (const float*)d_in[1];
    const float*         negatives = (const float*)d_in[2];
    const int*           pos_idx   = (const int*)d_in[3];
    const unsigned char* neg_mask  = (const unsigned char*)d_in[4];

    float* block_loss = (float*)d_ws;   // 2048 floats of scratch

    ntxent_main_kernel<<<B_, 256, 0, stream>>>(target, positives, negatives,
                                               pos_idx, neg_mask, block_loss);
    ntxent_reduce_kernel<<<1, 256, 0, stream>>>(block_loss, (float*)d_out);
}